// Ncc_59889023975763
// MI455X (gfx1250) — compile-verified
//
#include <hip/hip_runtime.h>
#include <hip/hip_bf16.h>

// Problem constants
#define B_    2048
#define P_    256
#define T_    365
#define D_    32
#define TD    11680            // T*D (divisible by 4 and 32)
#define TD4   2920             // TD/4
#define KAUG  12045            // TD + T (augmented K)
#define KPAD  12064            // 377 * 32, zero-padded
#define KC    32               // K-chunk staged in LDS
#define LDA   44               // LDS row pitch (floats): 16B-aligned float4s, 44r mod 64 conflict-free

typedef float v2f __attribute__((ext_vector_type(2)));
typedef float v8f __attribute__((ext_vector_type(8)));

// ---------------------------------------------------------------------------
// sq[p][t] = sum_d proto[p][t][d]^2  (float4 reads: 8 vectors per t-row)
__global__ __launch_bounds__(128)
void sq_kernel(const float4* __restrict__ proto4, float* __restrict__ sq) {
    const int p = blockIdx.x;
    for (int t = threadIdx.x; t < T_; t += 128) {
        const float4* row = proto4 + (long long)p * TD4 + t * (D_ / 4);
        float s = 0.0f;
#pragma unroll
        for (int d = 0; d < D_ / 4; ++d) {
            float4 v = row[d];
            s += v.x * v.x + v.y * v.y + v.z * v.z + v.w * v.w;
        }
        sq[p * T_ + t] = s;
    }
}

// ---------------------------------------------------------------------------
// term_x[b] = sum_{t,d} mask[b][t] * x[b][t][d]^2  (float4 reads)
__global__ __launch_bounds__(256)
void termx_kernel(const float4* __restrict__ x4, const float* __restrict__ mask,
                  float* __restrict__ term_x) {
    __shared__ float red[256];
    const int b = blockIdx.x;
    float s = 0.0f;
    for (int i = threadIdx.x; i < TD4; i += 256) {
        float4 xv = x4[(long long)b * TD4 + i];
        float  m  = mask[b * T_ + (i >> 3)];     // (4*i)>>5
        s += m * (xv.x * xv.x + xv.y * xv.y + xv.z * xv.z + xv.w * xv.w);
    }
    red[threadIdx.x] = s;
    __syncthreads();
    for (int off = 128; off > 0; off >>= 1) {
        if (threadIdx.x < off) red[threadIdx.x] += red[threadIdx.x + off];
        __syncthreads();
    }
    if (threadIdx.x == 0) term_x[b] = red[0];
}

// ---------------------------------------------------------------------------
// dist[b][p] = term_x[b] + A'[b,:] . B'[p,:]
//   A'[b,k<TD]    = -2*mask[b,k/32]*x[b,k]   A'[b,TD+t] = mask[b,t]
//   B'[p,k<TD]    = proto[p,k]               B'[p,TD+t] = sq[p,t]
// One workgroup (8 waves) per 16 rows of b; each wave computes 16x32 of dist
// via two fp32 WMMA accumulators. Tiles staged in LDS with float4 traffic.
__global__ __launch_bounds__(256)
void gemm_dist_kernel(const float4* __restrict__ x4, const float* __restrict__ mask,
                      const float4* __restrict__ proto4, const float* __restrict__ sq,
                      const float* __restrict__ term_x, float* __restrict__ dist) {
    __shared__ __attribute__((aligned(16))) float ldsA[16 * LDA];   //  2.8 KB
    __shared__ __attribute__((aligned(16))) float ldsB[P_ * LDA];   // 45.1 KB

    const int tid    = threadIdx.x;
    const int lane   = tid & 31;
    const int wave   = tid >> 5;
    const int m0     = blockIdx.x * 16;     // base row in B dimension
    const int n_base = wave * 32;           // 32 prototype columns per wave

    const int r    = lane & 15;             // row/col within 16
    const int half = (lane >> 4) << 1;      // K sub-offset: 0 (lanes 0-15) or 2 (16-31)

    v8f acc0 = {};
    v8f acc1 = {};

    for (int kc = 0; kc < KPAD; kc += KC) {
        __syncthreads();
        // Stage A' tile: 16 x KC as 128 float4s (mask & -2 fused; kg>>5 uniform per vec)
        for (int i = tid; i < 16 * (KC / 4); i += 256) {
            int kl4 = (i & 7) << 2;
            int m   = i >> 3;
            int kg  = kc + kl4;
            int b   = m0 + m;
            float4 v;
            if (kg < TD) {
                float  mf = -2.0f * mask[b * T_ + (kg >> 5)];
                float4 xv = x4[(long long)b * TD4 + (kg >> 2)];
                v = make_float4(mf * xv.x, mf * xv.y, mf * xv.z, mf * xv.w);
            } else {
                float e[4];
#pragma unroll
                for (int j = 0; j < 4; ++j) {
                    int k = kg + j;
                    e[j] = (k < KAUG) ? mask[b * T_ + (k - TD)] : 0.0f;
                }
                v = make_float4(e[0], e[1], e[2], e[3]);
            }
            *(float4*)(&ldsA[m * LDA + kl4]) = v;
        }
        // Stage B' tile: 256 x KC as 2048 float4s
        for (int i = tid; i < P_ * (KC / 4); i += 256) {
            int kl4 = (i & 7) << 2;
            int p   = i >> 3;
            int kg  = kc + kl4;
            float4 v;
            if (kg < TD) {
                v = proto4[(long long)p * TD4 + (kg >> 2)];
            } else {
                float e[4];
#pragma unroll
                for (int j = 0; j < 4; ++j) {
                    int k = kg + j;
                    e[j] = (k < KAUG) ? sq[p * T_ + (k - TD)] : 0.0f;
                }
                v = make_float4(e[0], e[1], e[2], e[3]);
            }
            *(float4*)(&ldsB[p * LDA + kl4]) = v;
        }
        __syncthreads();

#pragma unroll
        for (int kk = 0; kk < KC; kk += 4) {
            // A fragment (16x4): lane r holds row M=r, VGPR0/1 = K (half, half+1)
            v2f a, b0, b1;
            a.x  = ldsA[r * LDA + kk + half];
            a.y  = ldsA[r * LDA + kk + half + 1];
            // B fragments (4x16): lane r holds col N=r at the same K pair
            b0.x = ldsB[(n_base + r) * LDA + kk + half];
            b0.y = ldsB[(n_base + r) * LDA + kk + half + 1];
            b1.x = ldsB[(n_base + 16 + r) * LDA + kk + half];
            b1.y = ldsB[(n_base + 16 + r) * LDA + kk + half + 1];
            acc0 = __builtin_amdgcn_wmma_f32_16x16x4_f32(false, a, false, b0,
                                                         (short)0, acc0, false, false);
            acc1 = __builtin_amdgcn_wmma_f32_16x16x4_f32(false, a, false, b1,
                                                         (short)0, acc1, false, false);
        }
    }

    // C/D layout: VGPR v -> M = v + (lane>=16 ? 8 : 0), N = lane%16
    const int moff = (lane >> 4) * 8;
#pragma unroll
    for (int v = 0; v < 8; ++v) {
        int M = v + moff;
        float tx = term_x[m0 + M];
        dist[(m0 + M) * P_ + n_base + r]      = acc0[v] + tx;
        dist[(m0 + M) * P_ + n_base + 16 + r] = acc1[v] + tx;
    }
}

// ---------------------------------------------------------------------------
// argmin over P=256 per row b; low-index tie-break (matches jnp.argmin).
__global__ __launch_bounds__(256)
void argmin_kernel(const float* __restrict__ dist, int* __restrict__ idx,
                   float* __restrict__ out_idx) {
    __shared__ float vmin[256];
    __shared__ int   imin[256];
    const int b = blockIdx.x;
    const int p = threadIdx.x;
    vmin[p] = dist[b * P_ + p];
    imin[p] = p;
    __syncthreads();
    for (int off = 128; off > 0; off >>= 1) {
        if (p < off) {
            float v2 = vmin[p + off];
            if (v2 < vmin[p] || (v2 == vmin[p] && imin[p + off] < imin[p])) {
                vmin[p] = v2; imin[p] = imin[p + off];
            }
        }
        __syncthreads();
    }
    if (p == 0) { idx[b] = imin[0]; out_idx[b] = (float)imin[0]; }
}

// ---------------------------------------------------------------------------
// Codebook gather + pass-throughs, float4 (label scalar).
// Flat float4 work: [gather N4][copy-x N4][mask M4] + [label B_]
__global__ __launch_bounds__(256)
void epilogue_kernel(const float4* __restrict__ x4, const int* __restrict__ label,
                     const float4* __restrict__ mask4, const float4* __restrict__ proto4,
                     const int* __restrict__ idx, float* __restrict__ out) {
    const unsigned N4      = (unsigned)B_ * TD4;                       // 5,980,160
    const unsigned M4      = ((unsigned)B_ * T_) >> 2;                 //   186,880
    const unsigned o_lab   = 2u * (unsigned)B_ * TD + (unsigned)B_ * P_ + B_; // float idx
    const unsigned o_mask4 = (o_lab + B_) >> 2;
    const unsigned total   = 2u * N4 + M4 + B_;
    float4* out4 = (float4*)out;
    for (unsigned i = blockIdx.x * 256u + threadIdx.x; i < total;
         i += gridDim.x * 256u) {
        if (i < N4) {
            unsigned b  = i / TD4;
            unsigned rr = i - b * TD4;
            out4[i] = proto4[(unsigned long long)idx[b] * TD4 + rr];   // output_seq
        } else if (i < 2u * N4) {
            out4[i] = x4[i - N4];                                      // input_seq
        } else if (i < 2u * N4 + M4) {
            unsigned j = i - 2u * N4;
            out4[o_mask4 + j] = mask4[j];                              // mask
        } else {
            unsigned b = i - (2u * N4 + M4);
            out[o_lab + b] = (float)label[b];                          // label
        }
    }
}

// ---------------------------------------------------------------------------
extern "C" void kernel_launch(void* const* d_in, const int* in_sizes, int n_in,
                              void* d_out, int out_size, void* d_ws, size_t ws_size,
                              hipStream_t stream) {
    const float*  x      = (const float*) d_in[0];
    const int*    label  = (const int*)   d_in[1];
    const float*  mask   = (const float*) d_in[2];
    const float*  proto  = (const float*) d_in[3];
    const float4* x4     = (const float4*)d_in[0];
    const float4* mask4  = (const float4*)d_in[2];
    const float4* proto4 = (const float4*)d_in[3];
    float* out = (float*)d_out;
    (void)x;

    // workspace: sq[P*T] | term_x[B] | idx[B]
    float* sq     = (float*)d_ws;
    float* term_x = sq + (size_t)P_ * T_;
    int*   idx    = (int*)(term_x + B_);

    // d_out layout (floats): out_seq[B*TD] | in_seq[B*TD] | dist[B*P] | idx[B] | label[B] | mask[B*T]
    float* dist    = out + 2LL * B_ * TD;
    float* out_idx = dist + (long long)B_ * P_;

    sq_kernel   <<<P_, 128, 0, stream>>>(proto4, sq);
    termx_kernel<<<B_, 256, 0, stream>>>(x4, mask, term_x);
    gemm_dist_kernel<<<B_ / 16, 256, 0, stream>>>(x4, mask, proto4, sq, term_x, dist);
    argmin_kernel<<<B_, 256, 0, stream>>>(dist, idx, out_idx);

    unsigned total = 2u * (unsigned)B_ * TD4 + (((unsigned)B_ * T_) >> 2) + B_;
    epilogue_kernel<<<(total + 255u) / 256u, 256, 0, stream>>>(x4, label, mask4, proto4, idx, out);
}